// HierarchicalSoftmax_3298534884003
// MI455X (gfx1250) — compile-verified
//
#include <hip/hip_runtime.h>
#include <hip/hip_bf16.h>
#include <math.h>

typedef float v2f __attribute__((ext_vector_type(2)));
typedef float v8f __attribute__((ext_vector_type(8)));

#define V_VOCAB 10000
#define H_DIM   512
#define D_PATH  16
#define WAVES_PER_BLOCK 8          // 256 threads = 8 wave32 = 8 samples/block
#define KSTEPS  (H_DIM / 4)        // 128 WMMA 16x16x4 steps over K=512

// One wave computes one sample's 16 logits with V_WMMA_F32_16X16X4_F32,
// then masked BCE + per-wave reduction. Partial sums go to d_ws.
__global__ __launch_bounds__(256) void hs_loss_kernel(
    const float* __restrict__ hidden,        // [B, 512]
    const int*   __restrict__ target_words,  // [B]
    const float* __restrict__ weights,       // [V, 16, 512]
    const float* __restrict__ codes,         // [V, 16]
    const int*   __restrict__ lengths,       // [V]
    float*       __restrict__ partials)      // [gridDim.x]
{
    const int tid  = threadIdx.x;
    const int lane = tid & 31;
    const int wid  = tid >> 5;
    const int sample = blockIdx.x * WAVES_PER_BLOCK + wid;

    const int word = target_words[sample];
    const float* Wp = weights + (size_t)word * (D_PATH * H_DIM);
    const float* hp = hidden  + (size_t)sample * H_DIM;

    const int m = lane & 15;   // A-matrix row M  (lanes 0-15 / 16-31 both map M=lane%16)
    const int g = lane >> 4;   // K sub-pair: lanes 0-15 hold K=0,1; lanes 16-31 hold K=2,3

    // A fragment source: row m of the gathered weight tile, columns 4t+2g, 4t+2g+1
    const float* wrow = Wp + m * H_DIM + 2 * g;
    // B fragment source: hidden broadcast into all 16 columns (all columns of D identical)
    const float* hrow = hp + 2 * g;

    v8f acc = {0.f, 0.f, 0.f, 0.f, 0.f, 0.f, 0.f, 0.f};

#pragma unroll 8
    for (int t = 0; t < KSTEPS; ++t) {
        v2f a = *(const v2f*)(wrow + 4 * t);   // global_load_b64, contiguous per lane
        v2f b = *(const v2f*)(hrow + 4 * t);   // broadcast load within half-wave
        // 8 args: (neg_a, A, neg_b, B, c_mod, C, reuse_a, reuse_b)
        acc = __builtin_amdgcn_wmma_f32_16x16x4_f32(
            false, a, false, b, (short)0, acc, false, false);
    }

    // C/D layout: lane holds rows d = 8*g + v for VGPR v = 0..7 (every column equal).
    const float* cp = codes + (size_t)word * D_PATH + 8 * g;
    const int   L  = lengths[word];

    float s = 0.f;
#pragma unroll
    for (int v = 0; v < 8; ++v) {
        const int   d = 8 * g + v;
        const float x = acc[v];
        const float t = cp[v];
        // stable BCE-with-logits: max(x,0) - x*t + log1p(exp(-|x|))
        const float bce = fmaxf(x, 0.f) - x * t + log1pf(expf(-fabsf(x)));
        if (d < L) s += bce;
    }
    // All 16 columns carry identical logits -> keep only column 0 of each half-wave.
    if (m != 0) s = 0.f;

    // Wave32 reduction (lanes 0 and 16 carry d=0..7 and d=8..15 sums).
    for (int off = 16; off > 0; off >>= 1)
        s += __shfl_xor(s, off, 32);

    __shared__ float red[WAVES_PER_BLOCK];
    if (lane == 0) red[wid] = s / (float)L;   // mean over the huffman path
    __syncthreads();

    if (tid == 0) {
        float tsum = 0.f;
        for (int i = 0; i < WAVES_PER_BLOCK; ++i) tsum += red[i];
        partials[blockIdx.x] = tsum;
    }
}

// Deterministic final reduction: one block, fixed order.
__global__ __launch_bounds__(256) void hs_reduce_kernel(
    const float* __restrict__ partials, float* __restrict__ out,
    int n, float inv_batch)
{
    __shared__ float sm[256];
    float s = 0.f;
    for (int i = threadIdx.x; i < n; i += 256) s += partials[i];
    sm[threadIdx.x] = s;
    __syncthreads();
    for (int off = 128; off > 0; off >>= 1) {
        if (threadIdx.x < off) sm[threadIdx.x] += sm[threadIdx.x + off];
        __syncthreads();
    }
    if (threadIdx.x == 0) out[0] = sm[0] * inv_batch;
}

extern "C" void kernel_launch(void* const* d_in, const int* in_sizes, int n_in,
                              void* d_out, int out_size, void* d_ws, size_t ws_size,
                              hipStream_t stream) {
    const float* hidden       = (const float*)d_in[0];  // [B,512] f32
    const int*   target_words = (const int*)  d_in[1];  // [B] i32
    const float* weights      = (const float*)d_in[2];  // [V,16,512] f32
    const float* codes        = (const float*)d_in[3];  // [V,16] f32
    const int*   lengths      = (const int*)  d_in[4];  // [V] i32

    float* out      = (float*)d_out;
    float* partials = (float*)d_ws;

    const int B = in_sizes[1];                    // 8192
    const int nblocks = B / WAVES_PER_BLOCK;      // 1024

    hs_loss_kernel<<<nblocks, 256, 0, stream>>>(
        hidden, target_words, weights, codes, lengths, partials);
    hs_reduce_kernel<<<1, 256, 0, stream>>>(
        partials, out, nblocks, 1.0f / (float)B);
}